// VC_MODEL_37512244363908
// MI455X (gfx1250) — compile-verified
//
#include <hip/hip_runtime.h>

typedef __attribute__((ext_vector_type(2))) float v2f;
typedef __attribute__((ext_vector_type(8))) float v8f;

__device__ __forceinline__ float lrelu(float v) { return v > 0.f ? v : 0.01f * v; }
__device__ __forceinline__ int imin(int a, int b) { return a < b ? a : b; }
__device__ __forceinline__ int imax(int a, int b) { return a > b ? a : b; }

// ---------------------------------------------------------------------------
// Implicit-GEMM conv1d on the fp32 WMMA path (V_WMMA_F32_16X16X4_F32).
//   Y[b, g*Mg+r, t] = act( bias + sum_{ic,kk} W[g*Mg+r, ic, kk] *
//                          X[b, g*Cg+ic, t*stride - pad + kk*dil] )
// K is a template parameter (1/3/4/512) so ka/K is constant math.
// One wave computes a 16(M) x 64(N) tile: four 16x16 WMMA accumulators
// sharing a single A fragment. 4 waves/block => block covers 16 x 256.
// Loads are branchless BY CONSTRUCTION: indices clamped in-range, value
// multiplied by a {0,1} mask float (strict FP forbids folding 0*x, so the
// compiler must issue the load unconditionally -- no exec-mask branches).
// All index math is 32-bit; the 64-bit widen happens once per access.
// Depth-1 software pipeline: chunk k+4 is in flight during chunk k's WMMAs.
// A-frag layout (16x4 f32): lane L holds {K=2*(L>>4), K+1} of row M=L&15.
// B-frag symmetric. D: VGPR v = row v+8*(L>>4), col L&15.
// ---------------------------------------------------------------------------
template <int K>
__global__ __launch_bounds__(128) void k_conv1d_wmma(
    const float* __restrict__ X, const float* __restrict__ W,
    const float* __restrict__ Bias, float* __restrict__ Y,
    int Bn, int Cin, int Tin, int Cout, int Tout,
    int stride, int pad, int dil, int groups, int act)
{
  const int Mg   = Cout / groups;
  const int Cg   = Cin / groups;
  const int Kred = Cg * K;
  const int g    = blockIdx.z;
  const int wave = threadIdx.x >> 5;
  const int lane = threadIdx.x & 31;
  const int mbase = blockIdx.y * 16;
  const int nbase = (blockIdx.x * 4 + wave) * 64;
  const int Ntot  = Bn * Tout;
  const int mi = lane & 15;
  const int kq = lane >> 4;  // 0 or 1

  const int arow = mbase + mi;
  const bool rowOK = arow < Mg;
  const float rowM = rowOK ? 1.f : 0.f;
  const float* Wrow = W + (size_t)(g * Mg + (rowOK ? arow : 0)) * Kred;

  int  bI[4], tS[4], tB[4];
  float colM[4];
  const float* XB[4];
#pragma unroll
  for (int j = 0; j < 4; j++) {
    const int nc = nbase + 16 * j + mi;
    const bool ok = nc < Ntot;
    colM[j] = ok ? 1.f : 0.f;
    const int bb = ok ? nc / Tout : 0;
    const int tt = nc - bb * Tout;
    bI[j] = bb; tS[j] = tt; tB[j] = tt * stride - pad;
    XB[j] = X + ((size_t)bb * Cin + (size_t)g * Cg) * Tin;
  }

  __builtin_prefetch(Wrow, 0, 1);          // global_prefetch_b8
  __builtin_prefetch(XB[0], 0, 1);

  // Branchless fragment loaders: clamp index, unconditional load, mask-mul.
  auto loadA = [&](int k0) -> v2f {
    const int ka = k0 + 2 * kq;
    const float a0 = Wrow[imin(ka,     Kred - 1)];
    const float a1 = Wrow[imin(ka + 1, Kred - 1)];
    v2f av;
    av.x = a0 * (ka     < Kred ? rowM : 0.f);
    av.y = a1 * (ka + 1 < Kred ? rowM : 0.f);
    return av;
  };
  auto loadB = [&](int k0, int j) -> v2f {
    const int ka = k0 + 2 * kq;
    const int ic0 = ka / K,       kk0 = ka - ic0 * K;
    const int ic1 = (ka + 1) / K, kk1 = (ka + 1) - ic1 * K;
    const int ti0 = tB[j] + kk0 * dil;
    const int ti1 = tB[j] + kk1 * dil;
    const int idx0 = imin(ic0, Cg - 1) * Tin + imin(imax(ti0, 0), Tin - 1);
    const int idx1 = imin(ic1, Cg - 1) * Tin + imin(imax(ti1, 0), Tin - 1);
    const float v0 = XB[j][idx0];   // 32-bit index, widened once
    const float v1 = XB[j][idx1];
    v2f bv;
    bv.x = v0 * ((ka     < Kred && (unsigned)ti0 < (unsigned)Tin) ? colM[j] : 0.f);
    bv.y = v1 * ((ka + 1 < Kred && (unsigned)ti1 < (unsigned)Tin) ? colM[j] : 0.f);
    return bv;
  };

  v8f acc[4];
#pragma unroll
  for (int j = 0; j < 4; j++) acc[j] = (v8f){0.f,0.f,0.f,0.f,0.f,0.f,0.f,0.f};

  v2f av = loadA(0);
  v2f bv[4];
#pragma unroll
  for (int j = 0; j < 4; j++) bv[j] = loadB(0, j);

#pragma unroll 2
  for (int k0 = 0; k0 < Kred; k0 += 4) {
    const int kn = k0 + 4;                 // next chunk (masked past Kred)
    v2f avn = loadA(kn);
    v2f bvn[4];
#pragma unroll
    for (int j = 0; j < 4; j++) bvn[j] = loadB(kn, j);
#pragma unroll
    for (int j = 0; j < 4; j++)
      acc[j] = __builtin_amdgcn_wmma_f32_16x16x4_f32(false, av, false, bv[j],
                                                     (short)0, acc[j], false, false);
    av = avn;
#pragma unroll
    for (int j = 0; j < 4; j++) bv[j] = bvn[j];
  }

#pragma unroll
  for (int v = 0; v < 8; ++v) {
    const int r = mbase + v + 8 * kq;
    if (r < Mg) {
      const float bias = Bias[g * Mg + r];
      const size_t rowoff = (size_t)g * Mg + r;
#pragma unroll
      for (int j = 0; j < 4; j++) {
        if (nbase + 16 * j + mi < Ntot) {
          float val = acc[j][v] + bias;
          if (act) val = lrelu(val);
          Y[((size_t)bI[j] * Cout + rowoff) * Tout + tS[j]] = val;
        }
      }
    }
  }
}

// ---------------- GroupNorm(G) + leaky-relu, one block per (group, batch) ---
__global__ void k_groupnorm_lrelu(const float* __restrict__ X,
                                  const float* __restrict__ gw,
                                  const float* __restrict__ gb,
                                  float* __restrict__ Y, int C, int T, int G)
{
  const int g = blockIdx.x, b = blockIdx.y;
  const int Cg = C / G;
  const size_t base = ((size_t)b * C + (size_t)g * Cg) * T;
  const int n = Cg * T;
  __shared__ float s1[256], s2[256];
  float sum = 0.f, sq = 0.f;
  for (int i = threadIdx.x; i < n; i += 256) {
    float v = X[base + i];
    sum += v; sq += v * v;
  }
  s1[threadIdx.x] = sum; s2[threadIdx.x] = sq;
  __syncthreads();
  for (int st = 128; st > 0; st >>= 1) {
    if (threadIdx.x < st) {
      s1[threadIdx.x] += s1[threadIdx.x + st];
      s2[threadIdx.x] += s2[threadIdx.x + st];
    }
    __syncthreads();
  }
  const float mu  = s1[0] / n;
  const float var = s2[0] / n - mu * mu;
  const float inv = rsqrtf(var + 1e-5f);
  for (int i = threadIdx.x; i < n; i += 256) {
    int c = g * Cg + i / T;
    float v = (X[base + i] - mu) * inv * gw[c] + gb[c];
    Y[base + i] = lrelu(v);
  }
}

// ---- per-(b,c): mean over T and 1/(||x-mu||_2 + 1e-4) ----------------------
__global__ void k_rowstats(const float* __restrict__ X, float* __restrict__ mean,
                           float* __restrict__ inv, int C, int T)
{
  const int c = blockIdx.x, b = blockIdx.y;
  const float* row = X + ((size_t)b * C + c) * T;
  __shared__ float s[256];
  float sum = 0.f;
  for (int i = threadIdx.x; i < T; i += 256) sum += row[i];
  s[threadIdx.x] = sum; __syncthreads();
  for (int st = 128; st > 0; st >>= 1) {
    if (threadIdx.x < st) s[threadIdx.x] += s[threadIdx.x + st];
    __syncthreads();
  }
  const float mu = s[0] / T;
  __syncthreads();
  float sq = 0.f;
  for (int i = threadIdx.x; i < T; i += 256) { float v = row[i] - mu; sq += v * v; }
  s[threadIdx.x] = sq; __syncthreads();
  for (int st = 128; st > 0; st >>= 1) {
    if (threadIdx.x < st) s[threadIdx.x] += s[threadIdx.x + st];
    __syncthreads();
  }
  if (threadIdx.x == 0) {
    mean[b * C + c] = mu;
    inv[b * C + c]  = 1.f / (sqrtf(s[0]) + 1e-4f);
  }
}

__global__ void k_normalize(const float* X, const float* mean, const float* inv,
                            float* Y, int T, int n)
{
  int i = blockIdx.x * blockDim.x + threadIdx.x;
  if (i >= n) return;
  int bc = i / T;
  Y[i] = (X[i] - mean[bc]) * inv[bc];
}

__global__ void k_mean_T(const float* X, float* Y, int n_rows, int T)
{
  int i = blockIdx.x * blockDim.x + threadIdx.x;
  if (i >= n_rows) return;
  const float* row = X + (size_t)i * T;
  float s = 0.f;
  for (int t = 0; t < T; t++) s += row[t];
  Y[i] = s / T;
}

// ---- _quantize_simple: per (b,t) column of Xn (B,d,T), argmin over 64 codes.
__global__ void k_quant_simple(const float* __restrict__ w, const float* Xn,
                               float* Q, float* __restrict__ ssq,
                               int Bn, int d, int T, int ncb)
{
  int idx = blockIdx.x * blockDim.x + threadIdx.x;
  if (idx >= Bn * T) return;
  int b = idx / T, t = idx - b * T;
  float f[40];
  for (int c = 0; c < d; c++) f[c] = Xn[((size_t)b * d + c) * T + t];
  int best = 0; float bestd = 3.4e38f;
  for (int n = 0; n < ncb; n++) {
    const float* wn = w + (size_t)n * d;
    float dist = 0.f;
    for (int c = 0; c < d; c++) { float df = f[c] - wn[c]; dist += df * df; }
    if (dist < bestd) { bestd = dist; best = n; }
  }
  const float* wb = w + (size_t)best * d;
  float s = 0.f;
  for (int c = 0; c < d; c++) {
    float q = wb[c]; float df = q - f[c]; s += df * df;
    Q[((size_t)b * d + c) * T + t] = q;
  }
  ssq[idx] = s;
}

// ---- codebook build: cb[b,n,d] = normalize_d(emb[n,d,:] . sp3[b,:]) + sp2[b,d]
__global__ void k_make_cb(const float* __restrict__ emb, const float* __restrict__ sp3,
                          const float* __restrict__ sp2, float* __restrict__ cb,
                          int n, int d, int Bn)
{
  int i = blockIdx.x * blockDim.x + threadIdx.x;
  if (i >= n * Bn) return;
  int ni = i / Bn, b = i - ni * Bn;
  float v[40]; float ss = 0.f;
  for (int dd = 0; dd < d; dd++) {
    const float* er = emb + ((size_t)ni * d + dd) * d;
    float acc = 0.f;
    for (int e = 0; e < d; e++) acc += er[e] * sp3[b * d + e];
    v[dd] = acc; ss += acc * acc;
  }
  float inv = 1.f / sqrtf(ss);
  for (int dd = 0; dd < d; dd++)
    cb[((size_t)b * n + ni) * d + dd] = v[dd] * inv + sp2[b * d + dd];
}

// ---- _quantize: f = Xn[b,:,t] + sp1[b,:]; argmin_n |f - cb[b,n,:]|^2 --------
__global__ void k_quant_cb(const float* __restrict__ cb, const float* __restrict__ Xn,
                           const float* __restrict__ sp1, float* __restrict__ Q,
                           float* __restrict__ ssq, int Bn, int n, int d, int T)
{
  int idx = blockIdx.x * blockDim.x + threadIdx.x;
  if (idx >= Bn * T) return;
  int b = idx / T, t = idx - b * T;
  float f[40];
  for (int c = 0; c < d; c++) f[c] = Xn[((size_t)b * d + c) * T + t] + sp1[b * d + c];
  int best = 0; float bestd = 3.4e38f;
  const float* cbb = cb + (size_t)b * n * d;
  for (int ni = 0; ni < n; ni++) {
    const float* row = cbb + (size_t)ni * d;
    float dist = 0.f;
    for (int c = 0; c < d; c++) { float df = f[c] - row[c]; dist += df * df; }
    if (dist < bestd) { bestd = dist; best = ni; }
  }
  const float* row = cbb + (size_t)best * d;
  float s = 0.f;
  for (int c = 0; c < d; c++) {
    float q = row[c]; float df = q - f[c]; s += df * df;
    Q[((size_t)b * d + c) * T + t] = q;
  }
  ssq[idx] = s;
}

// ---- deterministic single-block reductions into the diff scalar ------------
__global__ void k_zero1(float* p) { p[0] = 0.f; }

__global__ void k_reduce_add_scaled(const float* __restrict__ src, int n,
                                    float scale, float* acc)
{
  __shared__ float s[256];
  float sum = 0.f;
  for (int i = threadIdx.x; i < n; i += 256) sum += src[i];
  s[threadIdx.x] = sum; __syncthreads();
  for (int st = 128; st > 0; st >>= 1) {
    if (threadIdx.x < st) s[threadIdx.x] += s[threadIdx.x + st];
    __syncthreads();
  }
  if (threadIdx.x == 0) acc[0] += s[0] * scale;
}

__global__ void k_reduce_diff_mean(const float* __restrict__ a, const float* __restrict__ b,
                                   int n, float scale, float* acc)
{
  __shared__ float s[256];
  float sum = 0.f;
  for (int i = threadIdx.x; i < n; i += 256) sum += a[i] - b[i];
  s[threadIdx.x] = sum; __syncthreads();
  for (int st = 128; st > 0; st >>= 1) {
    if (threadIdx.x < st) s[threadIdx.x] += s[threadIdx.x + st];
    __syncthreads();
  }
  if (threadIdx.x == 0) acc[0] += s[0] * scale;
}

// ---- elementwise helpers ---------------------------------------------------
__global__ void k_copy(float* dst, const float* src, int n) {
  int i = blockIdx.x * blockDim.x + threadIdx.x;
  if (i < n) dst[i] = src[i];
}
__global__ void k_add2(float* dst, const float* a, const float* b, int n) {
  int i = blockIdx.x * blockDim.x + threadIdx.x;
  if (i < n) dst[i] = a[i] + b[i];
}
__global__ void k_addip(float* dst, const float* src, int n) {
  int i = blockIdx.x * blockDim.x + threadIdx.x;
  if (i < n) dst[i] += src[i];
}
__global__ void k_repeat2(const float* X, float* Y, int T, int n_in) {
  int i = blockIdx.x * blockDim.x + threadIdx.x;
  if (i >= n_in) return;
  int row = i / T, t = i - row * T;
  float v = X[i];
  Y[(size_t)row * 2 * T + 2 * t]     = v;
  Y[(size_t)row * 2 * T + 2 * t + 1] = v;
}
__global__ void k_cat2(const float* x, const float* r, float* dst,
                       int d, int T, int n_in) {
  int i = blockIdx.x * blockDim.x + threadIdx.x;
  if (i >= n_in) return;
  int b = i / (d * T), rem = i - b * d * T;
  int c = rem / T, t = rem - c * T;
  float xv = x[i];
  dst[((size_t)b * 2 * d + c) * T + t]     = xv;
  dst[((size_t)b * 2 * d + d + c) * T + t] = xv + r[i];
}
__global__ void k_gather_rows(const float* src, const int* perm, float* dst,
                              int Bn, int d) {
  int i = blockIdx.x * blockDim.x + threadIdx.x;
  if (i >= Bn * d) return;
  int b = i / d, c = i - b * d;
  dst[i] = src[(size_t)perm[b] * d + c];
}

// ---------------------------------------------------------------------------
extern "C" void kernel_launch(void* const* d_in, const int* in_sizes, int n_in,
                              void* d_out, int out_size, void* d_ws, size_t ws_size,
                              hipStream_t stream)
{
  (void)in_sizes; (void)n_in; (void)out_size; (void)ws_size;
  const float* X0  = (const float*)d_in[0];
  const int*  perm = (const int*)d_in[1];
  float* out = (float*)d_out;

  // ---- param leaf indexing (jax tree_flatten: sorted dict keys) -----------
  auto F = [&](int i) -> const float* { return (const float*)d_in[i]; };
  struct GB { const float *c1b, *c1w, *c2b, *c2w, *gnb, *gnw; };
  GB blocksP[3], refineP[3], resP[3];
  int pi = 2;
  GB* kinds[3] = {blocksP, refineP, resP};           // 'blocks','refine','res'
  for (int k = 0; k < 3; k++)
    for (int s = 0; s < 3; s++) {
      GB& g = kinds[k][s];
      g.c1b = F(pi++); g.c1w = F(pi++); g.c2b = F(pi++); g.c2w = F(pi++);
      g.gnb = F(pi++); g.gnw = F(pi++);
    }
  struct EncP { const float *c1b, *c1w, *c2b, *c2w; } encp[3];
  for (int s = 0; s < 3; s++) { encp[s].c1b = F(pi++); encp[s].c1w = F(pi++);
                                encp[s].c2b = F(pi++); encp[s].c2w = F(pi++); }
  const float* qsw[3];    for (int s = 0; s < 3; s++) qsw[s]    = F(pi++);
  const float* quantw[3]; for (int s = 0; s < 3; s++) quantw[s] = F(pi++);
  struct SpkP { const float *c1b,*c1w,*c2b,*c2w,*c3b,*c3w,
                            *d1b,*d1w,*d2b,*d2w,*d3b,*d3w,*dnb,*dnw; } spkp[3];
  for (int s = 0; s < 3; s++) {
    SpkP& p = spkp[s];
    p.c1b=F(pi++); p.c1w=F(pi++); p.c2b=F(pi++); p.c2w=F(pi++);
    p.c3b=F(pi++); p.c3w=F(pi++); p.d1b=F(pi++); p.d1w=F(pi++);
    p.d2b=F(pi++); p.d2w=F(pi++); p.d3b=F(pi++); p.d3w=F(pi++);
    p.dnb=F(pi++); p.dnw=F(pi++);
  }

  // ---- output layout -------------------------------------------------------
  const int D1_0 = 0, D1_1 = 10240, D1_2 = 51200;
  const int D2_0 = 215040, D2_1 = 225280, D2_2 = 266240;
  const int EB0 = 430080, EB1 = 471040, EB2 = 481280, DIFF = 483840;
  float* diffp = out + DIFF;

  // ---- workspace bump allocator -------------------------------------------
  size_t wsoff = 0;
  auto wsf = [&](size_t nf) -> float* {
    float* p = (float*)((char*)d_ws + wsoff);
    wsoff += ((nf * sizeof(float) + 255) / 256) * 256;
    return p;
  };
  float* bufH  = wsf(4 * 512 * 512);
  float* bufX  = wsf(4 * 40 * 512);
  float* bufX2 = wsf(4 * 40 * 512);
  float* bufG  = wsf(4 * 40 * 512);
  float* h1    = wsf(4 * 128 * 256);
  float* h2    = wsf(4 * 128 * 256);
  float* xst   = wsf(4 * 40 * 256);
  float* xn0   = wsf(4 * 40 * 256);
  float* xn1   = wsf(4 * 20 * 128);
  float* xn2   = wsf(4 * 10 * 64);
  float* meanb = wsf(4 * 64);
  float* invb  = wsf(4 * 64);
  float* hv    = wsf(4 * 128);
  float* hv2   = wsf(4 * 128);
  float* sd[3][3];
  for (int s = 0; s < 3; s++) for (int j = 0; j < 3; j++) sd[s][j] = wsf(4 * 64);
  float* sdp1 = wsf(4 * 64);
  float* sdp2 = wsf(4 * 64);
  float* cb   = wsf(4 * 128 * 40);
  float* ssq  = wsf(4 * 512);
  float* ebc0 = wsf(4 * 40 * 256);
  float* ebc1 = wsf(4 * 20 * 128);
  float* ebc2 = wsf(4 * 10 * 64);

  auto conv = [&](const float* Xp, const float* Wp, const float* Bp, float* Yp,
                  int Bn, int Cin, int Tin, int Cout, int Tout,
                  int K, int stride, int pad, int dil, int groups, int act) {
    int Mg = Cout / groups, Ntot = Bn * Tout;
    dim3 grid((Ntot + 255) / 256, (Mg + 15) / 16, groups);
    switch (K) {
      case 1:
        k_conv1d_wmma<1><<<grid, 128, 0, stream>>>(Xp, Wp, Bp, Yp, Bn, Cin, Tin,
            Cout, Tout, stride, pad, dil, groups, act); break;
      case 3:
        k_conv1d_wmma<3><<<grid, 128, 0, stream>>>(Xp, Wp, Bp, Yp, Bn, Cin, Tin,
            Cout, Tout, stride, pad, dil, groups, act); break;
      case 4:
        k_conv1d_wmma<4><<<grid, 128, 0, stream>>>(Xp, Wp, Bp, Yp, Bn, Cin, Tin,
            Cout, Tout, stride, pad, dil, groups, act); break;
      default:
        k_conv1d_wmma<512><<<grid, 128, 0, stream>>>(Xp, Wp, Bp, Yp, Bn, Cin, Tin,
            Cout, Tout, stride, pad, dil, groups, act); break;
    }
  };
  auto gblock_run = [&](const GB& p, const float* xin, float* outp, int d, int T) {
    conv(xin, p.c1w, p.c1b, bufH, 4, d, T, 512, T, 3, 1, 1, 1, 1, 0);
    k_groupnorm_lrelu<<<dim3(4, 4), 256, 0, stream>>>(bufH, p.gnw, p.gnb, bufH, 512, T, 4);
    conv(bufH, p.c2w, p.c2b, outp, 4, 512, T, d, T, 3, 1, 1, 1, 1, 0);
  };

  k_zero1<<<1, 1, 0, stream>>>(diffp);

  // ============================ ENCODER =====================================
  const int dims[3] = {40, 20, 10}, Ts[3] = {256, 128, 64};
  const int cis[3] = {80, 40, 20}, gsz[3] = {4, 2, 1}, ncb[3] = {32, 64, 128};
  float* xns[3]  = {xn0, xn1, xn2};
  float* encb[3] = {out + EB0, out + EB1, out + EB2};
  const float* xcur = X0; int Tin = 512;
  const int pads = (512 - 1) * 8 / 2;  // 2044

  for (int i = 0; i < 3; i++) {
    const int d = dims[i], T = Ts[i], g = gsz[i];
    conv(xcur, encp[i].c1w, encp[i].c1b, bufH, 4, cis[i], Tin, 512, T, 4, 2, 1, 1, 1, 1);
    conv(bufH, encp[i].c2w, encp[i].c2b, xst, 4, 512, T, d, T, 3, 1, 1, 1, 1, 0);
    // speaker block
    const SpkP& sp = spkp[i];
    conv(xst, sp.c1w, sp.c1b, h1, 4, d, T, 128, T, 512, 1, pads, 8, g, 0);
    conv(h1, sp.c2w, sp.c2b, h2, 4, 128, T, 128, T, 512, 1, pads, 8, g, 1);
    conv(h2, sp.c3w, sp.c3b, h1, 4, 128, T, 128, T, 3, 1, 1, 1, 1, 0);
    k_mean_T<<<2, 256, 0, stream>>>(h1, hv, 4 * 128, T);
    conv(hv,  sp.dnw, sp.dnb, hv2,      4, 128, 1, 128, 1, 1, 1, 0, 1, 1, 1);
    conv(hv2, sp.d1w, sp.d1b, sd[i][0], 4, 128, 1, d,   1, 1, 1, 0, 1, 1, 0);
    conv(hv2, sp.d2w, sp.d2b, sd[i][1], 4, 128, 1, d,   1, 1, 1, 0, 1, 1, 0);
    conv(hv2, sp.d3w, sp.d3b, sd[i][2], 4, 128, 1, d,   1, 1, 1, 0, 1, 1, 0);
    // normalize: xn = (x - mean_T) / (||x-mean||_2 + 1e-4)
    k_rowstats<<<dim3(d, 4), 256, 0, stream>>>(xst, meanb, invb, d, T);
    const int n = 4 * d * T;
    k_normalize<<<(n + 255) / 256, 256, 0, stream>>>(xst, meanb, invb, xns[i], T, n);
    // quantize_simple (in-place on xn) + ds
    k_quant_simple<<<(4 * T + 255) / 256, 256, 0, stream>>>(qsw[i], xns[i], xns[i],
                                                            ssq, 4, d, T, 64);
    k_reduce_add_scaled<<<1, 256, 0, stream>>>(ssq, 4 * T, 1.f / (4.f * T * d), diffp);
    // speaker-conditioned codebook quantize + dq  (sp[1]=d2, sp[2]=d3)
    k_make_cb<<<(ncb[i] * 4 + 255) / 256, 256, 0, stream>>>(quantw[i], sd[i][2],
                                                            sd[i][1], cb, ncb[i], d, 4);
    k_quant_cb<<<(4 * T + 255) / 256, 256, 0, stream>>>(cb, xns[i], sd[i][1],
                                                        encb[i], ssq, 4, ncb[i], d, T);
    k_reduce_add_scaled<<<1, 256, 0, stream>>>(ssq, 4 * T, 1.f / (4.f * T * d), diffp);
    xcur = xst; Tin = T;
  }

  // ============================ DECODER =====================================
  auto decode = [&](const float* q0, const float* q1, const float* q2,
                    float* o0, float* o1, float* o2) {
    const float* qv[3] = {q0, q1, q2};
    float* ov[3] = {o0, o1, o2};
    const int dd[3] = {10, 20, 40}, TT[3] = {64, 128, 256};
    const float* xprev = nullptr;
    for (int i = 0; i < 3; i++) {
      const int d = dd[i], T = TT[i], T2 = 2 * T;
      const int n = 4 * d * T, n2 = 4 * d * T2;
      gblock_run(resP[i], qv[i], bufG, d, T);
      if (i == 0) k_copy<<<(n + 255) / 256, 256, 0, stream>>>(bufX, bufG, n);
      else        k_add2<<<(n + 255) / 256, 256, 0, stream>>>(bufX, xprev, bufG, n);
      k_repeat2<<<(n + 255) / 256, 256, 0, stream>>>(bufX, bufX2, T, n);
      gblock_run(blocksP[i], bufX2, bufG, d, T2);
      k_addip<<<(n2 + 255) / 256, 256, 0, stream>>>(bufX2, bufG, n2);
      gblock_run(refineP[i], bufX2, bufG, d, T2);
      k_cat2<<<(n2 + 255) / 256, 256, 0, stream>>>(bufX2, bufG, ov[i], d, T2, n2);
      xprev = ov[i];
    }
  };

  // dec1 from enc_b (reversed)
  decode(encb[2], encb[1], encb[0], out + D1_0, out + D1_1, out + D1_2);

  // enc_bc with batch-permuted speaker vectors
  float* ebc[3] = {ebc0, ebc1, ebc2};
  for (int i = 0; i < 3; i++) {
    const int d = dims[i], T = Ts[i];
    k_gather_rows<<<1, 256, 0, stream>>>(sd[i][1], perm, sdp1, 4, d);
    k_gather_rows<<<1, 256, 0, stream>>>(sd[i][2], perm, sdp2, 4, d);
    k_make_cb<<<(ncb[i] * 4 + 255) / 256, 256, 0, stream>>>(quantw[i], sdp2, sdp1,
                                                            cb, ncb[i], d, 4);
    k_quant_cb<<<(4 * T + 255) / 256, 256, 0, stream>>>(cb, xns[i], sdp1,
                                                        ebc[i], ssq, 4, ncb[i], d, T);
    k_reduce_diff_mean<<<1, 256, 0, stream>>>(encb[i], ebc[i], 4 * d * T,
                                              1.f / (4.f * d * T), diffp);
  }

  // dec2 from enc_bc (reversed)
  decode(ebc2, ebc1, ebc0, out + D2_0, out + D2_1, out + D2_2);
}